// HeteroGraphTransformer_74174085202175
// MI455X (gfx1250) — compile-verified
//
#include <hip/hip_runtime.h>
#include <hip/hip_bf16.h>
#include <math.h>

typedef __attribute__((ext_vector_type(2))) float v2f;
typedef __attribute__((ext_vector_type(8))) float v8f;

__device__ __forceinline__ float lrelu(float x) { return x > 0.f ? x : 0.2f * x; }

__device__ __forceinline__ void atomic_max_f32(float* addr, float val) {
  // monotonic bit-trick: works mixing signed-max (nonneg) with unsigned-min (neg)
  if (val >= 0.f)
    __hip_atomic_fetch_max((int*)addr, __float_as_int(val),
                           __ATOMIC_RELAXED, __HIP_MEMORY_SCOPE_AGENT);
  else
    __hip_atomic_fetch_min((unsigned int*)addr, __float_as_uint(val),
                           __ATOMIC_RELAXED, __HIP_MEMORY_SCOPE_AGENT);
}

__device__ __forceinline__ void atomic_add_f32(float* addr, float val) {
  __hip_atomic_fetch_add(addr, val, __ATOMIC_RELAXED, __HIP_MEMORY_SCOPE_AGENT);
}

// ---------------------------------------------------------------------------
// Precompute folded weight tables:
//   Vp[table][k][j] (64x16, j<4 valid): V[k,h] = sum_c W[k, h*64+c] * att[h,c]
//   Pc[r][j] (512x16, j<4 valid), r = layer*256 + h*64 + k:
//       P = (1/4) * sum_c W_layer[k, h*64+c] * W_out[layer*64+c, j]
//   cst[j] = b_cq . Wout_top[:,j] + b_ca . Wout_bot[:,j] + b_out[j]
// ---------------------------------------------------------------------------
__global__ __launch_bounds__(1024)
void prep_kernel(const float* __restrict__ Wsq, const float* __restrict__ Wdq,
                 const float* __restrict__ Asq, const float* __restrict__ Adq,
                 const float* __restrict__ Wsa, const float* __restrict__ Wda,
                 const float* __restrict__ Asa, const float* __restrict__ Ada,
                 const float* __restrict__ bq,  const float* __restrict__ ba,
                 const float* __restrict__ Wout, const float* __restrict__ bout,
                 float* __restrict__ Vp, float* __restrict__ Pc, float* __restrict__ cst) {
  const int t = threadIdx.x;
  for (int idx = t; idx < 4 * 64 * 16; idx += blockDim.x) {
    int table = idx >> 10, k = (idx >> 4) & 63, j = idx & 15;
    const float* W = (table == 0) ? Wsq : (table == 1) ? Wdq : (table == 2) ? Wsa : Wda;
    const float* A = (table == 0) ? Asq : (table == 1) ? Adq : (table == 2) ? Asa : Ada;
    float s = 0.f;
    if (j < 4)
      for (int c = 0; c < 64; ++c) s += W[k * 256 + j * 64 + c] * A[j * 64 + c];
    Vp[idx] = s;
  }
  for (int idx = t; idx < 512 * 16; idx += blockDim.x) {
    int r = idx >> 4, j = idx & 15;
    int layer = r >> 8, h = (r >> 6) & 3, k = r & 63;
    const float* W = layer ? Wsa : Wsq;
    float s = 0.f;
    if (j < 4) {
      for (int c = 0; c < 64; ++c)
        s += W[k * 256 + h * 64 + c] * Wout[(layer * 64 + c) * 4 + j];
      s *= 0.25f;
    }
    Pc[idx] = s;
  }
  if (t < 16) {
    float s = 0.f;
    if (t < 4) {
      for (int c = 0; c < 64; ++c)
        s += bq[c] * Wout[c * 4 + t] + ba[c] * Wout[(64 + c) * 4 + t];
      s += bout[t];
    }
    cst[t] = s;
  }
}

__global__ void init_kernel(float* __restrict__ zero_ptr, long nzero,
                            float* __restrict__ minf_ptr, long nminf) {
  const long stride = (long)gridDim.x * blockDim.x;
  const long t0 = (long)blockIdx.x * blockDim.x + threadIdx.x;
  for (long i = t0; i < nzero; i += stride) zero_ptr[i] = 0.f;
  for (long i = t0; i < nminf; i += stride) minf_ptr[i] = __int_as_float(0xff800000);
}

// ---------------------------------------------------------------------------
// Skinny GEMM via V_WMMA_F32_16X16X4_F32: out[N x 4] = x[N x 64] @ Vp[64 x 16pad]
// f32 A 16x4 layout: lane holds row (lane&15), K = (lane>>4)*2 + i
// f32 B 4x16 layout: lane holds col (lane&15), K = (lane>>4)*2 + i
// One wave computes 16 rows. Block = 8 waves = 128 rows.
// ---------------------------------------------------------------------------
__global__ void scores_kernel(const float* __restrict__ x, const float* __restrict__ Vp,
                              float* __restrict__ out, int N) {
  const int wave_g = (int)(((unsigned)blockIdx.x * blockDim.x + threadIdx.x) >> 5);
  const int lane = threadIdx.x & 31;
  const long base = (long)wave_g * 16;
  if (base >= N) return;                      // wave-uniform: EXEC stays all-ones
  const int m = lane & 15;
  const int ko = (lane >> 4) * 2;
  long row = base + m; if (row >= N) row = N - 1;
  const float* xr = x + row * 64;
  v8f acc = {};
#pragma unroll
  for (int kk = 0; kk < 64; kk += 4) {
    v2f a, b;
    a.x = xr[kk + ko];
    a.y = xr[kk + ko + 1];
    b.x = Vp[(kk + ko) * 16 + m];
    b.y = Vp[(kk + ko + 1) * 16 + m];
    acc = __builtin_amdgcn_wmma_f32_16x16x4_f32(false, a, false, b, (short)0, acc,
                                                false, false);
  }
  if (m < 4) {
    const long rb = base + ((lane >> 4) << 3);
#pragma unroll
    for (int v = 0; v < 8; ++v) {
      long r = rb + v;
      if (r < N) out[r * 4 + m] = acc[v];
    }
  }
}

// Edge pass 1: e = leaky_relu(a_s[src] + a_d[dst]); segment-max into m[dst]
__global__ void edge_pass1(const int* __restrict__ src, const int* __restrict__ dst,
                           const float* __restrict__ a_s, const float* __restrict__ a_d,
                           float* __restrict__ e, float* __restrict__ mbuf, int E) {
  const int i = blockIdx.x * blockDim.x + threadIdx.x;
  if (i >= E) return;
  const long s = src[i], d = dst[i];
  const float4 as = *(const float4*)(a_s + s * 4);
  const float4 ad = *(const float4*)(a_d + d * 4);
  float4 ev;
  ev.x = lrelu(as.x + ad.x);
  ev.y = lrelu(as.y + ad.y);
  ev.z = lrelu(as.z + ad.z);
  ev.w = lrelu(as.w + ad.w);
  *(float4*)(e + (long)i * 4) = ev;
  atomic_max_f32(mbuf + d * 4 + 0, ev.x);
  atomic_max_f32(mbuf + d * 4 + 1, ev.y);
  atomic_max_f32(mbuf + d * 4 + 2, ev.z);
  atomic_max_f32(mbuf + d * 4 + 3, ev.w);
}

// Edge pass 2: ex = exp(e - m[dst]) (in place); segment-sum into den[dst]
__global__ void edge_pass2(const int* __restrict__ dst, float* __restrict__ e,
                           const float* __restrict__ mbuf, float* __restrict__ den, int E) {
  const int i = blockIdx.x * blockDim.x + threadIdx.x;
  if (i >= E) return;
  const long d = dst[i];
  float4 ev = *(const float4*)(e + (long)i * 4);
  const float4 mv = *(const float4*)(mbuf + d * 4);
  ev.x = __expf(ev.x - mv.x);
  ev.y = __expf(ev.y - mv.y);
  ev.z = __expf(ev.z - mv.z);
  ev.w = __expf(ev.w - mv.w);
  *(float4*)(e + (long)i * 4) = ev;
  atomic_add_f32(den + d * 4 + 0, ev.x);
  atomic_add_f32(den + d * 4 + 1, ev.y);
  atomic_add_f32(den + d * 4 + 2, ev.z);
  atomic_add_f32(den + d * 4 + 3, ev.w);
}

// Edge pass 3: z[dst, h, :] += alpha[e,h] * x[src, :].  64 threads per edge,
// thread t: h = t>>4, k0 = (t&15)*4 (one float4 of x, 4 L2-resident atomics).
__global__ void edge_pass3(const int* __restrict__ src, const int* __restrict__ dst,
                           const float* __restrict__ x, const float* __restrict__ ex,
                           const float* __restrict__ den, float* __restrict__ z, long E) {
  const long gt = (long)blockIdx.x * blockDim.x + threadIdx.x;
  const long ei = gt >> 6;
  if (ei >= E) return;
  const int t = (int)(gt & 63);
  const int h = t >> 4;
  const int k0 = (t & 15) * 4;
  const long s = src[ei], d = dst[ei];
  const float alpha = ex[ei * 4 + h] / (den[d * 4 + h] + 1e-16f);
  const float4 xv = *(const float4*)(x + s * 64 + k0);
  float* zp = z + d * 256 + h * 64 + k0;
  atomic_add_f32(zp + 0, alpha * xv.x);
  atomic_add_f32(zp + 1, alpha * xv.y);
  atomic_add_f32(zp + 2, alpha * xv.z);
  atomic_add_f32(zp + 3, alpha * xv.w);
}

// Final: out[N x 4] = [z_q | z_a][N x 512] @ Pc[512 x 16pad] + cst, via WMMA f32.
__global__ void final_kernel(const float* __restrict__ zq, const float* __restrict__ za,
                             const float* __restrict__ Pc, const float* __restrict__ cst,
                             float* __restrict__ out, int N) {
  const int wave_g = (int)(((unsigned)blockIdx.x * blockDim.x + threadIdx.x) >> 5);
  const int lane = threadIdx.x & 31;
  const long base = (long)wave_g * 16;
  if (base >= N) return;
  const int m = lane & 15;
  const int ko = (lane >> 4) * 2;
  long row = base + m; if (row >= N) row = N - 1;
  const float* zqr = zq + row * 256;
  const float* zar = za + row * 256;
  v8f acc = {};
#pragma unroll 8
  for (int kk = 0; kk < 512; kk += 4) {
    const int k0 = kk + ko;                      // k0 and k0+1 never cross 256
    const float* zr = (k0 < 256) ? zqr : zar;
    const int kL = k0 & 255;
    v2f a, b;
    a.x = zr[kL];
    a.y = zr[kL + 1];
    b.x = Pc[k0 * 16 + m];
    b.y = Pc[(k0 + 1) * 16 + m];
    acc = __builtin_amdgcn_wmma_f32_16x16x4_f32(false, a, false, b, (short)0, acc,
                                                false, false);
  }
  if (m < 4) {
    const float c0 = cst[m];
    const long rb = base + ((lane >> 4) << 3);
#pragma unroll
    for (int v = 0; v < 8; ++v) {
      long r = rb + v;
      if (r < N) out[r * 4 + m] = acc[v] + c0;
    }
  }
}

extern "C" void kernel_launch(void* const* d_in, const int* in_sizes, int n_in,
                              void* d_out, int out_size, void* d_ws, size_t ws_size,
                              hipStream_t stream) {
  const float* x_c = (const float*)d_in[0];
  const float* x_q = (const float*)d_in[1];
  const float* x_a = (const float*)d_in[2];
  const int* src_cq = (const int*)d_in[3];
  const int* dst_cq = (const int*)d_in[4];
  const int* src_ca = (const int*)d_in[5];
  const int* dst_ca = (const int*)d_in[6];
  const float* Wsq = (const float*)d_in[7];
  const float* Wdq = (const float*)d_in[8];
  const float* Asq = (const float*)d_in[9];
  const float* Adq = (const float*)d_in[10];
  const float* bq  = (const float*)d_in[11];
  const float* Wsa = (const float*)d_in[12];
  const float* Wda = (const float*)d_in[13];
  const float* Asa = (const float*)d_in[14];
  const float* Ada = (const float*)d_in[15];
  const float* ba  = (const float*)d_in[16];
  const float* Wout = (const float*)d_in[17];
  const float* bout = (const float*)d_in[18];

  const long Nc = in_sizes[0] / 64;
  const long Nq = in_sizes[1] / 64;
  const long Na = in_sizes[2] / 64;
  const long Ecq = in_sizes[3];
  const long Eca = in_sizes[5];

  float* w = (float*)d_ws;
  size_t off = 0;
  auto alloc = [&](size_t n) -> float* {
    float* p = w + off;
    off += (n + 15) & ~(size_t)15;    // keep 64B alignment for float4 access
    return p;
  };
  float* Vp_sq = alloc(64 * 16);      // 4 contiguous 64x16 tables (prep writes all)
  float* Vp_dq = alloc(64 * 16);
  float* Vp_sa = alloc(64 * 16);
  float* Vp_da = alloc(64 * 16);
  float* Pc    = alloc(512 * 16);
  float* cst   = alloc(16);
  float* as_cq = alloc(Nc * 4);
  float* as_ca = alloc(Nc * 4);
  float* ad_cq = alloc(Nq * 4);
  float* ad_ca = alloc(Na * 4);
  float* e_cq  = alloc(Ecq * 4);
  float* e_ca  = alloc(Eca * 4);
  float* m_q   = alloc(Nq * 4);       // -inf region start
  float* m_a   = alloc(Na * 4);
  float* den_q = alloc(Nq * 4);       // zero region start
  float* den_a = alloc(Na * 4);
  float* z_q   = alloc(Nq * 256);
  float* z_a   = alloc(Na * 256);
  (void)den_a; (void)Vp_dq; (void)Vp_sa; (void)Vp_da;
  if (off * sizeof(float) > ws_size) return;   // workspace too small: bail

  const long nminf = (long)(den_q - m_q);
  const long nzero = (long)((z_a + ((Na * 256 + 15) & ~(size_t)15)) - den_q);

  prep_kernel<<<1, 1024, 0, stream>>>(Wsq, Wdq, Asq, Adq, Wsa, Wda, Asa, Ada,
                                      bq, ba, Wout, bout, Vp_sq, Pc, cst);
  init_kernel<<<2048, 256, 0, stream>>>(den_q, nzero, m_q, nminf);

  const int gNc = (int)((Nc + 127) / 128);
  const int gNq = (int)((Nq + 127) / 128);
  const int gNa = (int)((Na + 127) / 128);
  scores_kernel<<<gNc, 256, 0, stream>>>(x_c, Vp_sq, as_cq, (int)Nc);
  scores_kernel<<<gNc, 256, 0, stream>>>(x_c, Vp_sa, as_ca, (int)Nc);
  scores_kernel<<<gNq, 256, 0, stream>>>(x_q, Vp_dq, ad_cq, (int)Nq);
  scores_kernel<<<gNa, 256, 0, stream>>>(x_a, Vp_da, ad_ca, (int)Na);

  const int gEcq = (int)((Ecq + 255) / 256);
  const int gEca = (int)((Eca + 255) / 256);
  edge_pass1<<<gEcq, 256, 0, stream>>>(src_cq, dst_cq, as_cq, ad_cq, e_cq, m_q, (int)Ecq);
  edge_pass1<<<gEca, 256, 0, stream>>>(src_ca, dst_ca, as_ca, ad_ca, e_ca, m_a, (int)Eca);
  edge_pass2<<<gEcq, 256, 0, stream>>>(dst_cq, e_cq, m_q, den_q, (int)Ecq);
  edge_pass2<<<gEca, 256, 0, stream>>>(dst_ca, e_ca, m_a, den_a, (int)Eca);

  edge_pass3<<<(int)((Ecq * 64 + 255) / 256), 256, 0, stream>>>(src_cq, dst_cq, x_c,
                                                                e_cq, den_q, z_q, Ecq);
  edge_pass3<<<(int)((Eca * 64 + 255) / 256), 256, 0, stream>>>(src_ca, dst_ca, x_c,
                                                                e_ca, den_a, z_a, Eca);

  final_kernel<<<gNq, 256, 0, stream>>>(z_q, z_a, Pc, cst, (float*)d_out, (int)Nq);
}